// PerlinNoise_24043226923343
// MI455X (gfx1250) — compile-verified
//
#include <hip/hip_runtime.h>

#define BLOCK 256

// ---- gfx1250 async global->LDS staging (feature-detected) -------------------
#if defined(__has_builtin)
# if __has_builtin(__builtin_amdgcn_global_load_async_to_lds_b32)
#  define HAVE_ASYNC_COPY 1
# endif
#endif
#ifndef HAVE_ASYNC_COPY
# define HAVE_ASYNC_COPY 0
#endif

typedef __attribute__((address_space(1))) int gint;  // global int
typedef __attribute__((address_space(3))) int lint;  // LDS int

__device__ __forceinline__ float fade_f(float t) {
    // 6t^5 - 15t^4 + 10t^3
    return t * t * t * (t * (t * 6.0f - 15.0f) + 10.0f);
}
__device__ __forceinline__ float lerp_f(float a, float b, float t) {
    return a + t * (b - a);
}

__global__ __launch_bounds__(BLOCK) void perlin3d_kernel(
    const float* __restrict__ gx, const float* __restrict__ gy,
    const float* __restrict__ gz, const int* __restrict__ perm,
    float* __restrict__ out, int n)
{
    // permL[j] = perm[j & 255]          (raw hash chain values, j in [0,512))
    // gidx[j]  = perm[j & 255] % 12     (gradient index for the final level)
    __shared__ int permL[512];
    __shared__ int gidx[512];

    const int tid  = blockIdx.x * BLOCK + threadIdx.x;
    const int base = tid * 4;

    // Issue global loads early so VMEM latency overlaps LDS table staging.
    float4 x4v = {0.f, 0.f, 0.f, 0.f};
    float4 y4v = x4v, z4v = x4v;
    const bool full = (base + 3 < n);
    if (full) {
        x4v = *(const float4*)(gx + base);
        y4v = *(const float4*)(gy + base);
        z4v = *(const float4*)(gz + base);
    }

    // ---- stage permutation table into LDS (duplicated to 512) ----
    {
        const int t = threadIdx.x;  // 256 threads, all lanes active
#if HAVE_ASYNC_COPY
        __builtin_amdgcn_global_load_async_to_lds_b32(
            (gint*)(perm + t), (lint*)&permL[t], 0, 0);
        __builtin_amdgcn_global_load_async_to_lds_b32(
            (gint*)(perm + t), (lint*)&permL[t + 256], 0, 0);
# if __has_builtin(__builtin_amdgcn_s_wait_asynccnt)
        __builtin_amdgcn_s_wait_asynccnt(0);
# else
        asm volatile("s_wait_asynccnt 0x0" ::: "memory");
# endif
#else
        const int p = perm[t];
        permL[t]       = p;
        permL[t + 256] = p;
#endif
    }
    __syncthreads();
    {
        const int t = threadIdx.x;
        const int m = permL[t] % 12;
        gidx[t]       = m;
        gidx[t + 256] = m;  // permL[t+256] == permL[t]
    }
    __syncthreads();

    if (base >= n) return;

    auto noise1 = [&](float x, float y, float z) -> float {
        const float fx = floorf(x), fy = floorf(y), fz = floorf(z);
        const int xi = ((int)fx) & 255;
        const int yi = ((int)fy) & 255;
        const int zi = ((int)fz) & 255;
        const float xf = x - fx, yf = y - fy, zf = z - fz;
        const float u = fade_f(xf), v = fade_f(yf), w = fade_f(zf);

        // shared hash chain: h(a,b,c) = perm[(perm[(perm[a]+b)%256]+c)%256]
        const int px  = permL[xi];
        const int px1 = permL[xi + 1];
        const int A   = permL[px  + yi];
        const int A1  = permL[px  + yi + 1];
        const int B   = permL[px1 + yi];
        const int B1  = permL[px1 + yi + 1];

        const int haaa = gidx[A  + zi];
        const int haab = gidx[A  + zi + 1];
        const int haba = gidx[A1 + zi];
        const int habb = gidx[A1 + zi + 1];
        const int hbaa = gidx[B  + zi];
        const int hbab = gidx[B  + zi + 1];
        const int hbba = gidx[B1 + zi];
        const int hbbb = gidx[B1 + zi + 1];

        const float xm = xf - 1.0f, ym = yf - 1.0f, zm = zf - 1.0f;

        // Branchless dot with grad3[h], h in [0,12):
        //   rows 0-3: (±1,±1,0)  rows 4-7: (±1,0,±1)  rows 8-11: (0,±1,±1)
        auto gdot = [](int h, float dx, float dy, float dz) -> float {
            float a = (h < 8) ? dx : dy;
            float b = (h < 4) ? dy : dz;
            a = (h & 1) ? -a : a;
            b = (h & 2) ? -b : b;
            return a + b;
        };

        const float daaa = gdot(haaa, xf, yf, zf);
        const float dbaa = gdot(hbaa, xm, yf, zf);
        const float daba = gdot(haba, xf, ym, zf);
        const float dbba = gdot(hbba, xm, ym, zf);
        const float daab = gdot(haab, xf, yf, zm);
        const float dbab = gdot(hbab, xm, yf, zm);
        const float dabb = gdot(habb, xf, ym, zm);
        const float dbbb = gdot(hbbb, xm, ym, zm);

        const float x1 = lerp_f(daaa, dbaa, u);
        const float x2 = lerp_f(daba, dbba, u);
        const float x3 = lerp_f(daab, dbab, u);
        const float x4 = lerp_f(dabb, dbbb, u);
        const float y1 = lerp_f(x1, x2, v);
        const float y2 = lerp_f(x3, x4, v);
        return lerp_f(y1, y2, w);
    };

    if (full) {
        float4 r;
        r.x = noise1(x4v.x, y4v.x, z4v.x);
        r.y = noise1(x4v.y, y4v.y, z4v.y);
        r.z = noise1(x4v.z, y4v.z, z4v.z);
        r.w = noise1(x4v.w, y4v.w, z4v.w);
        *(float4*)(out + base) = r;
    } else {
        for (int i = base; i < n; ++i)
            out[i] = noise1(gx[i], gy[i], gz[i]);
    }
}

extern "C" void kernel_launch(void* const* d_in, const int* in_sizes, int n_in,
                              void* d_out, int out_size, void* d_ws, size_t ws_size,
                              hipStream_t stream) {
    (void)n_in; (void)d_ws; (void)ws_size;
    const float* x     = (const float*)d_in[0];
    const float* y     = (const float*)d_in[1];
    const float* z     = (const float*)d_in[2];
    const int*   perm  = (const int*)d_in[3];
    // d_in[4] (grad3) is not needed: the gradient dot is computed analytically
    // from the fixed 12-entry table the reference hardcodes.
    float* out = (float*)d_out;

    const int n = out_size;                 // 32*512*512 = 8,388,608
    const int threads = (n + 3) / 4;        // 4 points per thread (float4 path)
    const int blocks  = (threads + BLOCK - 1) / BLOCK;
    perlin3d_kernel<<<blocks, BLOCK, 0, stream>>>(x, y, z, perm, out, n);
}